// Hybrik_54889682043718
// MI455X (gfx1250) — compile-verified
//
#include <hip/hip_runtime.h>
#include <math.h>

#define EPSF 1e-8f
#define NB 512
#define NV 6890
#define V3 (NV*3)        // 20670
#define NHC 79           // 10 + 46 + 20 + 3 collapsed head outputs
#define HID 1024
#define KPD 207

typedef __attribute__((ext_vector_type(2))) float v2f;
typedef __attribute__((ext_vector_type(8))) float v8f;

__constant__ int c_par[24]   = {-1,0,0,0,1,2,3,4,5,6,7,8,9,9,9,12,13,14,16,17,18,19,20,21};
__constant__ int c_child[24] = {3,4,5,6,7,8,9,10,11,12,-1,-1,15,16,17,-1,18,19,20,21,22,23,-1,-1};

// ---------------- small math helpers ----------------
__device__ __forceinline__ float dot3(const float* a, const float* b) {
  return a[0]*b[0] + a[1]*b[1] + a[2]*b[2];
}
__device__ __forceinline__ void cross3(const float* a, const float* b, float* o) {
  o[0] = a[1]*b[2] - a[2]*b[1];
  o[1] = a[2]*b[0] - a[0]*b[2];
  o[2] = a[0]*b[1] - a[1]*b[0];
}
__device__ __forceinline__ float norm3(const float* a) { return sqrtf(dot3(a,a)); }
__device__ __forceinline__ void rodrigues(const float* u, float c, float s, float* R) {
  float x = u[0], y = u[1], z = u[2];
  float K[9] = {0.f,-z,y,  z,0.f,-x,  -y,x,0.f};
  float K2[9];
  #pragma unroll
  for (int i = 0; i < 3; ++i)
    #pragma unroll
    for (int j = 0; j < 3; ++j) {
      float acc = 0.f;
      #pragma unroll
      for (int k = 0; k < 3; ++k) acc += K[i*3+k]*K[k*3+j];
      K2[i*3+j] = acc;
    }
  float t = 1.f - c;
  #pragma unroll
  for (int i = 0; i < 9; ++i)
    R[i] = ((i == 0 || i == 4 || i == 8) ? 1.f : 0.f) + s*K[i] + t*K2[i];
}
__device__ __forceinline__ void mm3(const float* A, const float* B, float* C) {
  #pragma unroll
  for (int i = 0; i < 3; ++i)
    #pragma unroll
    for (int j = 0; j < 3; ++j) {
      float acc = 0.f;
      #pragma unroll
      for (int k = 0; k < 3; ++k) acc += A[i*3+k]*B[k*3+j];
      C[i*3+j] = acc;
    }
}
__device__ __forceinline__ void mv3(const float* A, const float* v, float* o) {
  #pragma unroll
  for (int i = 0; i < 3; ++i) o[i] = A[i*3+0]*v[0] + A[i*3+1]*v[1] + A[i*3+2]*v[2];
}
__device__ __forceinline__ void mtv3(const float* A, const float* v, float* o) {
  #pragma unroll
  for (int i = 0; i < 3; ++i) o[i] = A[0*3+i]*v[0] + A[1*3+i]*v[1] + A[2*3+i]*v[2];
}

// ---------------- tiny kernels ----------------
__global__ void k_zero(float* p, int n) {
  int i = blockIdx.x*256 + threadIdx.x;
  if (i < n) p[i] = 0.f;
}

// Collapse the 4 linear heads: Wc(72x79) = W1@W2 per head, bc(79) = b1@W2+b2.
__global__ void k_collapse(
    const float* __restrict__ ws1, const float* __restrict__ ws2,
    const float* __restrict__ bs1, const float* __restrict__ bs2,
    const float* __restrict__ wp1, const float* __restrict__ wp2,
    const float* __restrict__ bp1, const float* __restrict__ bp2,
    const float* __restrict__ wl1, const float* __restrict__ wl2,
    const float* __restrict__ bl1, const float* __restrict__ bl2,
    const float* __restrict__ wt1, const float* __restrict__ wt2,
    const float* __restrict__ bt1, const float* __restrict__ bt2,
    float* __restrict__ Wc, float* __restrict__ bc)
{
  int idx = blockIdx.x*256 + threadIdx.x;
  if (idx >= 72*NHC + NHC) return;
  bool isB = (idx >= 72*NHC);
  int j = isB ? (idx - 72*NHC) : (idx % NHC);
  int i = isB ? 0 : (idx / NHC);
  const float *w1, *w2, *b1, *b2; int od, jl;
  if (j < 10)      { w1 = ws1; w2 = ws2; b1 = bs1; b2 = bs2; od = 10; jl = j;      }
  else if (j < 56) { w1 = wp1; w2 = wp2; b1 = bp1; b2 = bp2; od = 46; jl = j - 10; }
  else if (j < 76) { w1 = wl1; w2 = wl2; b1 = bl1; b2 = bl2; od = 20; jl = j - 56; }
  else             { w1 = wt1; w2 = wt2; b1 = bt1; b2 = bt2; od = 3;  jl = j - 76; }
  float acc = 0.f;
  if (!isB) {
    for (int k = 0; k < HID; ++k) acc += w1[i*HID + k] * w2[k*od + jl];
    Wc[i*NHC + j] = acc;
  } else {
    for (int k = 0; k < HID; ++k) acc += b1[k] * w2[k*od + jl];
    bc[j] = acc + b2[jl];
  }
}

// heads(512x79) = x(512x72) @ Wc + bc
__global__ void k_heads(const float* __restrict__ x, const float* __restrict__ Wc,
                        const float* __restrict__ bc, float* __restrict__ heads) {
  int idx = blockIdx.x*256 + threadIdx.x;
  if (idx >= NB*NHC) return;
  int b = idx / NHC, j = idx % NHC;
  float acc = bc[j];
  for (int k = 0; k < 72; ++k) acc += x[b*72 + k] * Wc[k*NHC + j];
  heads[idx] = acc;
}

// Jv(24x3) = J_reg @ v_template ; Jsd(24x3x10) = J_reg @ shapedirs
__global__ __launch_bounds__(256) void k_jreg(
    const float* __restrict__ Jr, const float* __restrict__ vt,
    const float* __restrict__ sd, float* __restrict__ Jv, float* __restrict__ Jsd)
{
  __shared__ float red[256];
  int id = blockIdx.x;
  int j, c, l = -1;
  if (id < 72) { j = id/3; c = id%3; }
  else { int r = id - 72; j = r/30; r %= 30; c = r/10; l = r%10; }
  float acc = 0.f;
  for (int v = threadIdx.x; v < NV; v += 256) {
    float w = Jr[j*NV + v];
    acc += w * (l < 0 ? vt[v*3 + c] : sd[(v*3 + c)*10 + l]);
  }
  red[threadIdx.x] = acc;
  __syncthreads();
  for (int st = 128; st > 0; st >>= 1) {
    if (threadIdx.x < st) red[threadIdx.x] += red[threadIdx.x + st];
    __syncthreads();
  }
  if (threadIdx.x == 0) {
    if (l < 0) Jv[j*3 + c] = red[0];
    else       Jsd[(j*3 + c)*10 + l] = red[0];
  }
}

// sdT(10, V3): sdT[l][m] = shapedirs[m*10+l]
__global__ void k_sdt(const float* __restrict__ sd, float* __restrict__ sdT) {
  int idx = blockIdx.x*256 + threadIdx.x;
  if (idx >= 10*V3) return;
  int l = idx / V3, m = idx % V3;
  sdT[idx] = sd[m*10 + l];
}

// ---------------- per-batch serial IK + FK ----------------
__global__ __launch_bounds__(64) void k_ik(
    const float* __restrict__ x, const float* __restrict__ heads,
    const float* __restrict__ Jv, const float* __restrict__ Jsd,
    float* __restrict__ pf, float* __restrict__ Arel, float* __restrict__ out_pose)
{
  int b = blockIdx.x*64 + threadIdx.x;
  if (b >= NB) return;
  const float* xb = x + b*72;
  const float* hb = heads + b*NHC;
  for (int k = 0; k < 72; ++k) out_pose[b*72 + k] = xb[k];   // pred_pose output

  float restJ[24][3], ps[24][3], relR[24][3], fp[24][3];
  for (int j = 0; j < 24; ++j)
    for (int c = 0; c < 3; ++c) {
      float a = Jv[j*3 + c];
      for (int l = 0; l < 10; ++l) a += Jsd[(j*3 + c)*10 + l] * hb[l];
      restJ[j][c] = a;
      ps[j][c] = 2.f * xb[j*3 + c];
    }
  for (int c = 0; c < 3; ++c) relR[0][c] = restJ[0][c];
  for (int j = 1; j < 24; ++j) {
    int p = c_par[j];
    for (int c = 0; c < 3; ++c) relR[j][c] = restJ[j][c] - restJ[p][c];
  }
  for (int j = 0; j < 24; ++j)
    for (int c = 0; c < 3; ++c) fp[j][c] = ps[j][c] - ps[0][c] + relR[0][c];

  // ---- pelvis orientation ----
  float rp1[3], rp2[3], rp3[3];
  for (int c = 0; c < 3; ++c) { rp1[c] = ps[1][c]-ps[0][c]; rp2[c] = ps[2][c]-ps[0][c]; rp3[c] = ps[3][c]-ps[0][c]; }
  float sf[3], sr[3], n;
  n = norm3(rp3) + EPSF;        for (int c = 0; c < 3; ++c) sf[c] = rp3[c]/n;
  n = norm3(relR[3]) + EPSF;    for (int c = 0; c < 3; ++c) sr[c] = relR[3][c]/n;
  float axis[3], u[3];
  cross3(sr, sf, axis);
  float s_ = norm3(axis), c_ = dot3(sr, sf);
  float inv = 1.f/(s_ + EPSF);  for (int c = 0; c < 3; ++c) u[c] = axis[c]*inv;
  float rot_spine[9]; rodrigues(u, c_, s_, rot_spine);
  float cf[3], cr[3], tmp3[3];
  for (int c = 0; c < 3; ++c) { cf[c] = 0.5f*(rp1[c]+rp2[c]); tmp3[c] = 0.5f*(relR[1][c]+relR[2][c]); }
  mv3(rot_spine, tmp3, cr);
  float d = dot3(cf, sf); for (int c = 0; c < 3; ++c) cf[c] -= d*sf[c];
  d = dot3(cr, sf);       for (int c = 0; c < 3; ++c) cr[c] -= d*sf[c];
  float nf = norm3(cf), nr = norm3(cr);
  cross3(cr, cf, axis);
  float an = norm3(axis), den = nr*nf + EPSF;
  c_ = dot3(cr, cf)/den; s_ = an/den;
  inv = 1.f/(an + EPSF); for (int c = 0; c < 3; ++c) u[c] = axis[c]*inv;
  float Rtmp[9]; rodrigues(u, c_, s_, Rtmp);

  float chain[24][9], rr[24][3];
  mm3(Rtmp, rot_spine, chain[0]);
  for (int c = 0; c < 3; ++c) rr[0][c] = relR[0][c];

  float* pfb = pf + b*KPD;
  int leaf_cnt = 0;
  for (int i = 1; i < 24; ++i) {
    int p = c_par[i], ch = c_child[i];
    float t3[3]; mv3(chain[p], relR[i], t3);
    for (int c = 0; c < 3; ++c) rr[i][c] = rr[p][c] + t3[c];
    float R[9];
    if (ch == -1) {
      const float* q = hb + 56 + leaf_cnt*4; leaf_cnt++;
      float qn = sqrtf(q[0]*q[0] + q[1]*q[1] + q[2]*q[2] + q[3]*q[3]) + EPSF;
      float w = q[0]/qn, qx = q[1]/qn, qy = q[2]/qn, qz = q[3]/qn;
      R[0] = 1.f-2.f*(qy*qy+qz*qz); R[1] = 2.f*(qx*qy-w*qz);       R[2] = 2.f*(qx*qz+w*qy);
      R[3] = 2.f*(qx*qy+w*qz);      R[4] = 1.f-2.f*(qx*qx+qz*qz);  R[5] = 2.f*(qy*qz-w*qx);
      R[6] = 2.f*(qx*qz-w*qy);      R[7] = 2.f*(qy*qz+w*qx);       R[8] = 1.f-2.f*(qx*qx+qy*qy);
    } else {
      float dv[3]; for (int c = 0; c < 3; ++c) dv[c] = fp[ch][c] - rr[i][c];
      float cfv[3]; mtv3(chain[p], dv, cfv);
      const float* crv = relR[ch];
      float nf2 = norm3(cfv), nr2 = norm3(crv);
      float ax[3]; cross3(crv, cfv, ax);
      float an2 = norm3(ax), den2 = nr2*nf2 + EPSF;
      float cc = dot3(crv, cfv)/den2, ss = an2/den2;
      float uu[3]; float iv = 1.f/(an2 + EPSF);
      for (int c = 0; c < 3; ++c) uu[c] = ax[c]*iv;
      float Rs[9]; rodrigues(uu, cc, ss, Rs);
      float ph0 = hb[10 + (i-1)*2], ph1 = hb[10 + (i-1)*2 + 1];
      float pn = sqrtf(ph0*ph0 + ph1*ph1) + EPSF; ph0 /= pn; ph1 /= pn;
      float ut[3]; iv = 1.f/(nr2 + EPSF);
      for (int c = 0; c < 3; ++c) ut[c] = crv[c]*iv;
      float Rt[9]; rodrigues(ut, ph0, ph1, Rt);
      mm3(Rs, Rt, R);
    }
    mm3(chain[p], R, chain[i]);
    for (int rc = 0; rc < 9; ++rc)
      pfb[(i-1)*9 + rc] = R[rc] - ((rc==0||rc==4||rc==8) ? 1.f : 0.f);
  }
  // A_rel: rotation = chain[j], translation = rr[j] - chain[j] @ restJ[j]
  float* Ab = Arel + b*288;
  for (int j = 0; j < 24; ++j) {
    float tv[3]; mv3(chain[j], restJ[j], tv);
    for (int r = 0; r < 3; ++r) {
      Ab[j*12 + r*4 + 0] = chain[j][r*3 + 0];
      Ab[j*12 + r*4 + 1] = chain[j][r*3 + 1];
      Ab[j*12 + r*4 + 2] = chain[j][r*3 + 2];
      Ab[j*12 + r*4 + 3] = rr[j][r] - tv[r];
    }
  }
}

// ---------------- fused blend GEMM (WMMA fp32 16x16x4) ----------------
// mesh[b, v*3+c] = v_template + shapedirs·betas + pose_feature·posedirs
// Block: 64 batch rows staged in LDS, 64 output columns; 4 waves x 16 cols;
// each wave holds 4 M-subtile accumulators so every B load feeds 4 WMMAs.
// Pipelined over the 51 full posedirs chunks; clamped epilogues for the
// posedirs tail (A[207]=0 pad), shapedirs (K=10, A pad 0) and v_template.
__global__ __launch_bounds__(128) void k_poseblend(
    const float* __restrict__ pf, const float* __restrict__ heads,
    const float* __restrict__ sdT, const float* __restrict__ vt,
    const float* __restrict__ posedirs, float* __restrict__ mesh)
{
  __shared__ float Alds[64][228];   // padded stride for bank-conflict relief
  const int m0 = blockIdx.y * 64;
  for (int idx = threadIdx.x; idx < 64*224; idx += 128) {
    int row = idx / 224, kk = idx % 224;
    float v = 0.f;
    if (kk < KPD)                    v = pf[(m0+row)*KPD + kk];
    else if (kk >= 208 && kk < 218)  v = heads[(m0+row)*NHC + (kk-208)];
    else if (kk == 220)              v = 1.f;
    Alds[row][kk] = v;
  }
  __syncthreads();

  const int wave = threadIdx.x >> 5;
  const int lane = threadIdx.x & 31;
  const int half = lane >> 4;
  const int l16  = lane & 15;
  const int n0   = blockIdx.x * 64 + wave * 16;
  const int col  = min(n0 + l16, V3 - 1);
  const float* bcol = posedirs + col;

  v8f acc[4];
  #pragma unroll
  for (int t = 0; t < 4; ++t) acc[t] = (v8f){};

  // --- pipelined main loop: posedirs chunks 0..50 (rows 0..203, no clamp) ---
  float bx = bcol[(size_t)(2*half)     * V3];
  float by = bcol[(size_t)(2*half + 1) * V3];
  for (int ck = 0; ck < 51; ++ck) {
    float nbx = 0.f, nby = 0.f;
    if (ck < 50) {                        // prefetch next chunk's B operand
      nbx = bcol[(size_t)(4*(ck+1) + 2*half)     * V3];
      nby = bcol[(size_t)(4*(ck+1) + 2*half + 1) * V3];
    }
    v2f bv = { bx, by };
    int ak = 4*ck + 2*half;
    #pragma unroll
    for (int t = 0; t < 4; ++t) {
      v2f a = { Alds[16*t + l16][ak], Alds[16*t + l16][ak + 1] };
      acc[t] = __builtin_amdgcn_wmma_f32_16x16x4_f32(false, a, false, bv,
                                                     (short)0, acc[t], false, false);
    }
    bx = nbx; by = nby;
  }
  // --- posedirs tail chunk 51: rows 204..207, clamp row to 206 (A[207]=0) ---
  {
    int k0 = 204 + 2*half;
    v2f bv = { bcol[(size_t)min(k0, 206)     * V3],
               bcol[(size_t)min(k0 + 1, 206) * V3] };
    #pragma unroll
    for (int t = 0; t < 4; ++t) {
      v2f a = { Alds[16*t + l16][k0], Alds[16*t + l16][k0 + 1] };
      acc[t] = __builtin_amdgcn_wmma_f32_16x16x4_f32(false, a, false, bv,
                                                     (short)0, acc[t], false, false);
    }
  }
  // --- shapedirs segment: K=10 at A offset 208 (A[218..219]=0) ---
  const float* scol = sdT + col;
  #pragma unroll
  for (int c2 = 0; c2 < 3; ++c2) {
    int k0 = 4*c2 + 2*half;
    v2f bv = { scol[(size_t)min(k0, 9)     * V3],
               scol[(size_t)min(k0 + 1, 9) * V3] };
    int ak = 208 + k0;
    #pragma unroll
    for (int t = 0; t < 4; ++t) {
      v2f a = { Alds[16*t + l16][ak], Alds[16*t + l16][ak + 1] };
      acc[t] = __builtin_amdgcn_wmma_f32_16x16x4_f32(false, a, false, bv,
                                                     (short)0, acc[t], false, false);
    }
  }
  // --- v_template segment: K=1 at A offset 220 (A[221..223]=0) ---
  {
    float bvt = vt[col];
    v2f bv = { bvt, bvt };
    int ak = 220 + 2*half;
    #pragma unroll
    for (int t = 0; t < 4; ++t) {
      v2f a = { Alds[16*t + l16][ak], Alds[16*t + l16][ak + 1] };
      acc[t] = __builtin_amdgcn_wmma_f32_16x16x4_f32(false, a, false, bv,
                                                     (short)0, acc[t], false, false);
    }
  }

  int coln = n0 + l16;
  if (coln < V3) {
    #pragma unroll
    for (int t = 0; t < 4; ++t)
      #pragma unroll
      for (int r = 0; r < 8; ++r)
        mesh[(size_t)(m0 + 16*t + r + 8*half) * V3 + coln] = acc[t][r];
  }
}

// ---------------- LBS skinning (fused, in-place on mesh) ----------------
__global__ __launch_bounds__(256) void k_skin(
    const float* __restrict__ lbs, const float* __restrict__ Arel,
    float* __restrict__ mesh)
{
  __shared__ float wsh[32][25];     // stride 25: conflict-free
  __shared__ float ash[8*288];
  const int v0 = blockIdx.x * 32;
  const int b0 = blockIdx.y * 8;
  for (int idx = threadIdx.x; idx < 32*24; idx += 256) {
    int v = idx / 24, j = idx % 24;
    int gv = min(v0 + v, NV - 1);
    wsh[v][j] = lbs[gv*24 + j];
  }
  for (int idx = threadIdx.x; idx < 8*288; idx += 256)
    ash[idx] = Arel[(size_t)(b0 + idx/288)*288 + (idx % 288)];
  __syncthreads();

  int vloc = threadIdx.x & 31, bloc = threadIdx.x >> 5;
  int v = v0 + vloc;
  if (v >= NV) return;
  int b = b0 + bloc;
  size_t base = (size_t)b * V3 + (size_t)v*3;
  float p0 = mesh[base], p1 = mesh[base+1], p2 = mesh[base+2];
  float T[12];
  #pragma unroll
  for (int e = 0; e < 12; ++e) T[e] = 0.f;
  const float* ab = &ash[bloc*288];
  #pragma unroll 4
  for (int j = 0; j < 24; ++j) {
    float wj = wsh[vloc][j];
    #pragma unroll
    for (int e = 0; e < 12; ++e) T[e] += wj * ab[j*12 + e];
  }
  mesh[base+0] = T[0]*p0 + T[1]*p1 + T[2]*p2  + T[3];
  mesh[base+1] = T[4]*p0 + T[5]*p1 + T[6]*p2  + T[7];
  mesh[base+2] = T[8]*p0 + T[9]*p1 + T[10]*p2 + T[11];
}

// root[b] = sum_v Jh36m[0,v] * verts[b,v]
__global__ __launch_bounds__(256) void k_root(
    const float* __restrict__ jh, const float* __restrict__ mesh,
    float* __restrict__ root)
{
  __shared__ float red[768];
  int b = blockIdx.y;
  int v = blockIdx.x*256 + threadIdx.x;
  float s0 = 0.f, s1 = 0.f, s2 = 0.f;
  if (v < NV) {
    float c = jh[v];  // row 0 of J_regressor_h36m
    size_t base = (size_t)b * V3 + (size_t)v*3;
    s0 = c*mesh[base]; s1 = c*mesh[base+1]; s2 = c*mesh[base+2];
  }
  red[threadIdx.x] = s0; red[256+threadIdx.x] = s1; red[512+threadIdx.x] = s2;
  __syncthreads();
  for (int st = 128; st > 0; st >>= 1) {
    if (threadIdx.x < st) {
      red[threadIdx.x]     += red[threadIdx.x + st];
      red[256+threadIdx.x] += red[256+threadIdx.x + st];
      red[512+threadIdx.x] += red[512+threadIdx.x + st];
    }
    __syncthreads();
  }
  if (threadIdx.x == 0) {
    atomicAdd(&root[b*3+0], red[0]);
    atomicAdd(&root[b*3+1], red[256]);
    atomicAdd(&root[b*3+2], red[512]);
  }
}

// mesh += pred_trans - root
__global__ void k_final(const float* __restrict__ heads, const float* __restrict__ root,
                        float* __restrict__ mesh) {
  int idx = blockIdx.x*256 + threadIdx.x;
  if (idx >= NB*V3) return;
  int b = idx / V3;
  int c = (idx % V3) % 3;
  mesh[idx] += heads[b*NHC + 76 + c] - root[b*3 + c];
}

// ---------------- launch ----------------
extern "C" void kernel_launch(void* const* d_in, const int* in_sizes, int n_in,
                              void* d_out, int out_size, void* d_ws, size_t ws_size,
                              hipStream_t stream) {
  const float* x    = (const float*)d_in[0];
  const float* ws1  = (const float*)d_in[1];
  const float* bs1  = (const float*)d_in[2];
  const float* ws2  = (const float*)d_in[3];
  const float* bs2  = (const float*)d_in[4];
  const float* wp1  = (const float*)d_in[5];
  const float* bp1  = (const float*)d_in[6];
  const float* wp2  = (const float*)d_in[7];
  const float* bp2  = (const float*)d_in[8];
  const float* wl1  = (const float*)d_in[9];
  const float* bl1  = (const float*)d_in[10];
  const float* wl2  = (const float*)d_in[11];
  const float* bl2  = (const float*)d_in[12];
  const float* wt1  = (const float*)d_in[13];
  const float* bt1  = (const float*)d_in[14];
  const float* wt2  = (const float*)d_in[15];
  const float* bt2  = (const float*)d_in[16];
  const float* vt   = (const float*)d_in[17];
  const float* sd   = (const float*)d_in[18];
  const float* pdir = (const float*)d_in[19];
  const float* Jr   = (const float*)d_in[20];
  const float* Jh   = (const float*)d_in[21];
  const float* lbs  = (const float*)d_in[22];

  float* out  = (float*)d_out;
  float* mesh = out + NB*72;
  float* ws   = (float*)d_ws;
  float* Wc    = ws + 0;        // 5688
  float* bc    = ws + 5696;     // 79
  float* heads = ws + 5824;     // 40448
  float* Jv    = ws + 46336;    // 72
  float* Jsd   = ws + 46464;    // 720
  float* pf    = ws + 47232;    // 105984
  float* Arel  = ws + 153216;   // 147456
  float* sdT   = ws + 300672;   // 206700
  float* root  = ws + 507392;   // 1536   (total ~2 MB)

  k_zero<<<(NB*3 + 255)/256, 256, 0, stream>>>(root, NB*3);
  k_collapse<<<(72*NHC + NHC + 255)/256, 256, 0, stream>>>(
      ws1, ws2, bs1, bs2, wp1, wp2, bp1, bp2,
      wl1, wl2, bl1, bl2, wt1, wt2, bt1, bt2, Wc, bc);
  k_heads<<<(NB*NHC + 255)/256, 256, 0, stream>>>(x, Wc, bc, heads);
  k_jreg<<<792, 256, 0, stream>>>(Jr, vt, sd, Jv, Jsd);
  k_sdt<<<(10*V3 + 255)/256, 256, 0, stream>>>(sd, sdT);
  k_ik<<<(NB + 63)/64, 64, 0, stream>>>(x, heads, Jv, Jsd, pf, Arel, out);

  dim3 gB((V3 + 63)/64, NB/64);
  k_poseblend<<<gB, 128, 0, stream>>>(pf, heads, sdT, vt, pdir, mesh);

  dim3 gS((NV + 31)/32, NB/8);
  k_skin<<<gS, 256, 0, stream>>>(lbs, Arel, mesh);

  dim3 gR((NV + 255)/256, NB);
  k_root<<<gR, 256, 0, stream>>>(Jh, mesh, root);

  k_final<<<(NB*V3 + 255)/256, 256, 0, stream>>>(heads, root, mesh);
}